// CrossCorrelationComputation_39264591020658
// MI455X (gfx1250) — compile-verified
//
#include <hip/hip_runtime.h>

// ---------------------------------------------------------------------------
// CrossCorrelationComputation for MI455X (gfx1250, wave32, WMMA)
//
// corr[q,s,p,k] = sum_c Qn[q,p,c] * Sn_shifted[s,c,p,k]
//               = Gram[p, p + offset_k]   (zero outside 14x14 border)
// where Gram = Qn[q] (196x640) x Sn[s]^T (640x196), computed with
// V_WMMA_F32_16X16X4_F32. Normalization scales are folded in (A rows scaled
// at LDS staging, B columns scaled at gather).
// ---------------------------------------------------------------------------

typedef float v2f __attribute__((ext_vector_type(2)));
typedef float v8f __attribute__((ext_vector_type(8)));

#define C_DIM 640
#define HW    196      // 14*14
#define SIDE  14
#define NS    25       // num support
#define NQ    75       // num query
#define KK    25       // 5*5 kernel offsets
#define EPSN  1e-8f

#define NTILES   13            // ceil(196/16) row tiles
#define BLOCK_T  (NTILES * 32) // 13 waves = 416 threads
#define AS_STRIDE 644          // 640 + 4: 644 % 64 == 4 -> conflict-free LDS
#define M_STRIDE  212          // 208 + 4: de-conflict half-wave row offset

// ---------------------------------------------------------------------------
// Kernel 1: inverse L2 norms over channels for every (image, position).
// Reads are coalesced over p (lane = position).
// ---------------------------------------------------------------------------
__global__ __launch_bounds__(224) void norm_kernel(
    const float* __restrict__ support,
    const float* __restrict__ query,
    float* __restrict__ inv_s,
    float* __restrict__ inv_q) {
  const int n = blockIdx.x;      // 0..99 : first NS are support, rest query
  const int p = threadIdx.x;     // position
  if (p >= HW) return;

  const float* base;
  float* out;
  if (n < NS) {
    base = support + (size_t)n * C_DIM * HW;
    out  = inv_s + n * HW;
  } else {
    base = query + (size_t)(n - NS) * C_DIM * HW;
    out  = inv_q + (n - NS) * HW;
  }

  float acc = 0.0f;
#pragma unroll 8
  for (int c = 0; c < C_DIM; ++c) {
    float v = base[c * HW + p];
    acc = fmaf(v, v, acc);
  }
  out[p] = 1.0f / fmaxf(sqrtf(acc), EPSN);
}

// ---------------------------------------------------------------------------
// Kernel 2: per (q, s, ptile): 16x196 Gram stripe via f32 WMMA, then gather
// the 25 shifted entries per row.
//   grid = (13, 25, 75), block = 416 threads (13 waves, wave w owns p' tile w)
// ---------------------------------------------------------------------------
__global__ __launch_bounds__(BLOCK_T) void gram_gather_kernel(
    const float* __restrict__ support,
    const float* __restrict__ query,
    const float* __restrict__ inv_s,
    const float* __restrict__ inv_q,
    float* __restrict__ out) {
  __shared__ float As[16 * AS_STRIDE];    // normalized query slab, [pl][c]
  __shared__ float Mrow[16 * M_STRIDE];   // 16 x 208 Gram stripe

  const int p0  = blockIdx.x * 16;   // query-position row tile base
  const int s   = blockIdx.y;
  const int q   = blockIdx.z;
  const int tid = threadIdx.x;

  // ---- Stage A slab into LDS: As[pl][c] = query[q][c][p0+pl] * inv_q ------
  // Global reads are 16-lane contiguous chunks over p; LDS stride 644 avoids
  // bank conflicts on the [pl][c] store.
  {
    const float* Qg = query + (size_t)q * C_DIM * HW;
    const float* iq = inv_q + q * HW;
    for (int idx = tid; idx < 16 * C_DIM; idx += BLOCK_T) {
      const int pl = idx & 15;
      const int c  = idx >> 4;
      const int p  = p0 + pl;
      float v = 0.0f;
      if (p < HW) v = Qg[c * HW + p] * iq[p];
      As[pl * AS_STRIDE + c] = v;
    }
  }
  __syncthreads();

  // ---- Per-wave 16x16 Gram tile: 160 x V_WMMA_F32_16X16X4_F32 -------------
  const int wave = tid >> 5;           // 0..12 -> p' tile
  const int lane = tid & 31;
  const int nl   = lane & 15;
  const int koff = (lane >> 4) << 1;   // 0 (lanes 0-15) or 2 (lanes 16-31)

  // ISA 16x4 f32 A layout: VGPR0 = {K0 | K2}, VGPR1 = {K1 | K3}
  // -> per lane the pair (c0+koff, c0+koff+1) is contiguous: ds_load_b64.
  const float* Arow = &As[nl * AS_STRIDE];

  // B column p' (clamped out-of-range columns produce garbage Gram entries
  // that are never gathered; clamping keeps loads in-bounds).
  int pc = wave * 16 + nl;
  if (pc >= HW) pc = HW - 1;
  const float* Bg = support + (size_t)s * C_DIM * HW + pc;

  v8f acc = {0.f, 0.f, 0.f, 0.f, 0.f, 0.f, 0.f, 0.f};
#pragma unroll 4
  for (int c0 = 0; c0 < C_DIM; c0 += 4) {
    v2f a = *(const v2f*)(Arow + c0 + koff);   // 8B aligned (even fp offset)
    v2f b;
    b.x = Bg[(c0 + koff) * HW];
    b.y = Bg[(c0 + koff + 1) * HW];
    acc = __builtin_amdgcn_wmma_f32_16x16x4_f32(
        /*neg_a=*/false, a, /*neg_b=*/false, b,
        /*c_mod=*/(short)0, acc, /*reuse_a=*/false, /*reuse_b=*/false);
  }

  // ---- Spill C tile to LDS. C layout: VGPR j -> row j (lanes 0-15) or
  //      row j+8 (lanes 16-31), column = lane%16. -------------------------
  {
    const int rbase = (lane >> 4) << 3;   // 0 or 8
    const int col   = wave * 16 + nl;     // 0..207
#pragma unroll
    for (int j = 0; j < 8; ++j) {
      Mrow[(rbase + j) * M_STRIDE + col] = acc[j];
    }
  }
  __syncthreads();

  // ---- Gather: out[q][s][p][k] = Gram[p, p+offset_k] * inv_s[p'] ----------
  if (tid < 16 * KK) {
    const int pl = tid / KK;
    const int k  = tid % KK;
    const int p  = p0 + pl;
    if (p < HW) {
      const int py = p / SIDE, px = p % SIDE;
      const int yy = py + (k / 5) - 2;
      const int xx = px + (k % 5) - 2;
      float val = 0.0f;
      if (yy >= 0 && yy < SIDE && xx >= 0 && xx < SIDE) {
        const int pp = yy * SIDE + xx;
        val = Mrow[pl * M_STRIDE + pp] * inv_s[s * HW + pp];
      }
      out[(((size_t)q * NS + s) * HW + p) * KK + k] = val;
    }
  }
}

// ---------------------------------------------------------------------------
extern "C" void kernel_launch(void* const* d_in, const int* in_sizes, int n_in,
                              void* d_out, int out_size, void* d_ws, size_t ws_size,
                              hipStream_t stream) {
  const float* support = (const float*)d_in[0];   // (25, 640, 14, 14)
  const float* query   = (const float*)d_in[1];   // (75, 640, 14, 14)
  float* inv_s = (float*)d_ws;                    // 25*196 floats
  float* inv_q = inv_s + NS * HW;                 // 75*196 floats
  float* outp  = (float*)d_out;                   // (75, 25, 196, 25)

  norm_kernel<<<dim3(NS + NQ), dim3(224), 0, stream>>>(
      support, query, inv_s, inv_q);

  gram_gather_kernel<<<dim3(NTILES, NS, NQ), dim3(BLOCK_T), 0, stream>>>(
      support, query, inv_s, inv_q, outp);
}